// GNNLayer_71648644432079
// MI455X (gfx1250) — compile-verified
//
#include <hip/hip_runtime.h>

// ---------------------------------------------------------------------------
// GNN layer, MI455X (gfx1250, wave32, WMMA + async global->LDS).
// Shapes: B=4, N=256, D=E=256.
//  cat = relu([start,end,edges] @ W_in + b_in)   -> split into P1(b,i)+P2(b,j)+edges@W3
//  coef = cat @ W_coef ; attn = softmax(aux+coef, axis=2)
//  out  = nodes + relu(sum_j attn*cat @ W_out + b_out)
// ---------------------------------------------------------------------------

typedef __attribute__((ext_vector_type(2))) float v2f;
typedef __attribute__((ext_vector_type(8))) float v8f;

#define WMMA_F32(a, b, c) \
  __builtin_amdgcn_wmma_f32_16x16x4_f32(false, (a), false, (b), (short)0, (c), false, false)

constexpr int Bc = 4, Nc = 256, Dc = 256, Ec = 256;
constexpr int LDSA_STRIDE = 260;  // 256 + 4 pad dwords: bank stride 4 -> conflict-free
constexpr int ROWS = 32;          // j-rows per workgroup in k_edge

// ---------------------------------------------------------------------------
// Kernel 0: P1 = nodes @ W_in[0:256] + b_in ; P2 = nodes @ W_in[256:512]
// One wave per 16x16 output tile; A fragment shared by both accumulators.
// ---------------------------------------------------------------------------
__global__ __launch_bounds__(32) void k_proj(const float* __restrict__ nodes,
                                             const float* __restrict__ W_in,
                                             const float* __restrict__ b_in,
                                             float* __restrict__ P1,
                                             float* __restrict__ P2) {
  const int rt = blockIdx.x;          // row tile over B*N = 1024 rows
  const int ct = blockIdx.y;          // col tile over 256 cols
  const int lane = threadIdx.x;
  const int mh = lane & 15;
  const int half = lane >> 4;
  const int koff = half * 2;
  const int r0 = rt * 16;
  const int c = ct * 16 + mh;

  const float* W1 = W_in;             // rows 0..255   of [768,256]
  const float* W2 = W_in + 256 * Dc;  // rows 256..511

  v8f acc1 = {};
  v8f acc2 = {};
  for (int k0 = 0; k0 < Dc; k0 += 4) {
    const int kk = k0 + koff;
    v2f a, b1, b2;
    a.x = nodes[(r0 + mh) * Dc + kk];
    a.y = nodes[(r0 + mh) * Dc + kk + 1];
    b1.x = W1[kk * Dc + c];
    b1.y = W1[(kk + 1) * Dc + c];
    b2.x = W2[kk * Dc + c];
    b2.y = W2[(kk + 1) * Dc + c];
    acc1 = WMMA_F32(a, b1, acc1);
    acc2 = WMMA_F32(a, b2, acc2);
  }
  const float bb = b_in[c];
#pragma unroll
  for (int r = 0; r < 8; ++r) {
    const int m = r + 8 * half;
    P1[(size_t)(r0 + m) * Dc + c] = acc1[r] + bb;
    P2[(size_t)(r0 + m) * Dc + c] = acc2[r];
  }
}

// ---------------------------------------------------------------------------
// Kernel 1: for each (b,i) and 32-row j-tile:
//   T[j,c]   = relu(edges[b,i,j,:] @ W3 + P1[b,i,c] + P2[b,j,c])  -> cat_out
//   logit[j] = aux[b,i,j] + T[j,:] . W_coef + b_coef              -> logits
// 128 threads (4 waves); A tile async-staged in LDS; each wave owns a 64-col
// slice and TWO 16-row tiles, so every B fragment feeds 2 WMMAs.
// ---------------------------------------------------------------------------
__global__ __launch_bounds__(128) void k_edge(const float* __restrict__ edges,
                                              const float* __restrict__ W_in,
                                              const float* __restrict__ P1,
                                              const float* __restrict__ P2,
                                              const float* __restrict__ W_coef,
                                              const float* __restrict__ b_coef,
                                              const float* __restrict__ aux,
                                              float* __restrict__ cat_out,
                                              float* __restrict__ logits) {
  __shared__ float sA[ROWS * LDSA_STRIDE];
  __shared__ float sCoef[ROWS];

  const int bi = blockIdx.x;  // b*N + i  (0..1023)
  const int jt = blockIdx.y;  // 32-row j tile (0..7)
  const int j0 = jt * ROWS;
  const int tid = threadIdx.x;
  const int lane = tid & 31;
  const int wave = tid >> 5;
  const int mh = lane & 15;
  const int half = lane >> 4;
  const int koff = half * 2;
  const int b = bi >> 8;  // N == 256

  // --- async DMA of the contiguous 32x256 edges tile into padded LDS
  // (GLOBAL_LOAD_ASYNC_TO_LDS_B128, tracked by ASYNCcnt; no VGPR data path)
  const float* Abase = edges + ((size_t)bi * Nc + j0) * Ec;
#pragma unroll
  for (int v = 0; v < 16; ++v) {
    const int q = tid + v * 128;  // float4 index 0..2047
    const int d = q * 4;          // dword index within tile
    const int m = d >> 8;
    const int e = d & 255;
    const float* gptr = Abase + d;
    const unsigned ldsOff = (unsigned)(uintptr_t)(&sA[m * LDSA_STRIDE + e]);
    asm volatile("global_load_async_to_lds_b128 %0, %1, off"
                 :
                 : "v"(ldsOff), "v"(gptr)
                 : "memory");
  }
  if (tid < ROWS) sCoef[tid] = 0.0f;
  asm volatile("s_wait_asynccnt 0x0" ::: "memory");
  __syncthreads();

  // --- GEMM: 2 x (16 rows) x 64 cols per wave, K = 256 in steps of 4
  const float* W3 = W_in + 512 * Dc;  // rows 512..767 of [768,256]
  const int cbase = wave * 64;
  v8f acc[2][4] = {};
  for (int k0 = 0; k0 < Ec; k0 += 4) {
    const int kk = k0 + koff;
    const float2 av0 = *reinterpret_cast<const float2*>(&sA[mh * LDSA_STRIDE + kk]);
    const float2 av1 = *reinterpret_cast<const float2*>(&sA[(16 + mh) * LDSA_STRIDE + kk]);
    v2f a0, a1;
    a0.x = av0.x; a0.y = av0.y;
    a1.x = av1.x; a1.y = av1.y;
#pragma unroll
    for (int t = 0; t < 4; ++t) {
      const int c = cbase + t * 16 + mh;
      v2f bf;
      bf.x = W3[kk * Dc + c];
      bf.y = W3[(kk + 1) * Dc + c];
      acc[0][t] = WMMA_F32(a0, bf, acc[0][t]);
      acc[1][t] = WMMA_F32(a1, bf, acc[1][t]);
    }
  }

  // --- epilogue: +P1+P2, relu, store cat, row-dot with W_coef
  const float* p1row = P1 + (size_t)bi * Dc;
  const float* p2base = P2 + (size_t)b * Nc * Dc;
#pragma unroll
  for (int t = 0; t < 4; ++t) {
    const int c = cbase + t * 16 + mh;
    const float p1 = p1row[c];
    const float wc = W_coef[c];
#pragma unroll
    for (int u = 0; u < 2; ++u) {
#pragma unroll
      for (int r = 0; r < 8; ++r) {
        const int mrow = u * 16 + r + 8 * half;
        const int j = j0 + mrow;
        float v = acc[u][t][r] + p1 + p2base[(size_t)j * Dc + c];
        v = fmaxf(v, 0.0f);
        cat_out[((size_t)bi * Nc + j) * Dc + c] = v;
        float pv = v * wc;
        pv += __shfl_xor(pv, 1, 16);
        pv += __shfl_xor(pv, 2, 16);
        pv += __shfl_xor(pv, 4, 16);
        pv += __shfl_xor(pv, 8, 16);
        if (mh == 0) atomicAdd(&sCoef[mrow], pv);
      }
    }
  }
  __syncthreads();

  if (tid < ROWS) {
    const int j = j0 + tid;
    logits[(size_t)bi * Nc + j] = aux[(size_t)bi * Nc + j] + sCoef[tid] + b_coef[0];
  }
}

// ---------------------------------------------------------------------------
// Kernel 2: per (b,i): attn = softmax(logits), resid[c] = sum_j attn[j]*cat[j,c]
// ---------------------------------------------------------------------------
__global__ __launch_bounds__(256) void k_softmax_resid(const float* __restrict__ logits,
                                                       const float* __restrict__ cat,
                                                       float* __restrict__ resid) {
  __shared__ float sAttn[256];
  __shared__ float sRed[256];
  const int bi = blockIdx.x;
  const int t = threadIdx.x;

  const float lg = logits[(size_t)bi * Nc + t];
  sRed[t] = lg;
  __syncthreads();
  for (int s = 128; s > 0; s >>= 1) {
    if (t < s) sRed[t] = fmaxf(sRed[t], sRed[t + s]);
    __syncthreads();
  }
  const float mx = sRed[0];
  __syncthreads();
  const float e = __expf(lg - mx);
  sRed[t] = e;
  __syncthreads();
  for (int s = 128; s > 0; s >>= 1) {
    if (t < s) sRed[t] += sRed[t + s];
    __syncthreads();
  }
  sAttn[t] = e * (1.0f / sRed[0]);
  __syncthreads();

  const float* catcol = cat + (size_t)bi * Nc * Dc + t;  // column t, stride Dc
  float acc = 0.0f;
  for (int j = 0; j < Nc; ++j) acc += sAttn[j] * catcol[(size_t)j * Dc];
  resid[(size_t)bi * Dc + t] = acc;
}

// ---------------------------------------------------------------------------
// Kernel 3: out = nodes + relu(resid @ W_out + b_out)
// ---------------------------------------------------------------------------
__global__ __launch_bounds__(32) void k_out(const float* __restrict__ resid,
                                            const float* __restrict__ W_out,
                                            const float* __restrict__ b_out,
                                            const float* __restrict__ nodes,
                                            float* __restrict__ out_nodes) {
  const int rt = blockIdx.x;
  const int ct = blockIdx.y;
  const int lane = threadIdx.x;
  const int mh = lane & 15;
  const int half = lane >> 4;
  const int koff = half * 2;
  const int r0 = rt * 16;
  const int c = ct * 16 + mh;

  v8f acc = {};
  for (int k0 = 0; k0 < Dc; k0 += 4) {
    const int kk = k0 + koff;
    v2f a, bf;
    a.x = resid[(r0 + mh) * Dc + kk];
    a.y = resid[(r0 + mh) * Dc + kk + 1];
    bf.x = W_out[kk * Dc + c];
    bf.y = W_out[(kk + 1) * Dc + c];
    acc = WMMA_F32(a, bf, acc);
  }
  const float bb = b_out[c];
#pragma unroll
  for (int r = 0; r < 8; ++r) {
    const int m = r + 8 * half;
    const float v = fmaxf(acc[r] + bb, 0.0f);
    out_nodes[(size_t)(r0 + m) * Dc + c] = nodes[(size_t)(r0 + m) * Dc + c] + v;
  }
}

// ---------------------------------------------------------------------------
extern "C" void kernel_launch(void* const* d_in, const int* in_sizes, int n_in,
                              void* d_out, int out_size, void* d_ws, size_t ws_size,
                              hipStream_t stream) {
  const float* nodes  = (const float*)d_in[0];
  const float* edges  = (const float*)d_in[1];
  const float* aux    = (const float*)d_in[2];
  // d_in[3] = nums (int scalar), unused
  const float* W_in   = (const float*)d_in[4];
  const float* b_in   = (const float*)d_in[5];
  const float* W_coef = (const float*)d_in[6];
  const float* b_coef = (const float*)d_in[7];
  const float* W_out  = (const float*)d_in[8];
  const float* b_out  = (const float*)d_in[9];

  float* out_nodes = (float*)d_out;                            // [B,N,D]
  float* out_cat   = (float*)d_out + (size_t)Bc * Nc * Dc;     // [B,N,N,D]

  float* P1     = (float*)d_ws;                                // [B*N, D]
  float* P2     = P1 + (size_t)Bc * Nc * Dc;                   // [B*N, D]
  float* logits = P2 + (size_t)Bc * Nc * Dc;                   // [B*N, N]
  float* resid  = logits + (size_t)Bc * Nc * Nc;               // [B*N, D]

  k_proj<<<dim3((Bc * Nc) / 16, Dc / 16), 32, 0, stream>>>(nodes, W_in, b_in, P1, P2);
  k_edge<<<dim3(Bc * Nc, Nc / ROWS), 128, 0, stream>>>(edges, W_in, P1, P2, W_coef, b_coef,
                                                       aux, out_cat, logits);
  k_softmax_resid<<<dim3(Bc * Nc), 256, 0, stream>>>(logits, out_cat, resid);
  k_out<<<dim3((Bc * Nc) / 16, Dc / 16), 32, 0, stream>>>(resid, W_out, b_out, nodes, out_nodes);
}